// ByteBitwiseFFN_7945689497941
// MI455X (gfx1250) — compile-verified
//
#include <hip/hip_runtime.h>
#include <stdint.h>

// ByteBitwiseFFN for MI455X (gfx1250).
// Memory-bound: 64 MiB in + 64 MiB out => ~5.8us floor at 23.3 TB/s.
// Tables replaced by integer ALU (and_table[a,b] == float(a&b), etc).
// CDNA5 async global->LDS b128 staging (ASYNCcnt), wave32 max-butterfly +
// ballot/ffs argmax (first-index semantics == jnp.argmax), b128 stores.

#define D_FEAT 128
#define ROWS_PER_BLOCK 16
#define THREADS 256

// Argmax over the 16 lanes of this half-wave. `v` is this lane's candidate,
// `shift` selects this half's bits of the wave32 ballot (0 or 16).
// Returns the SMALLEST lane index attaining the maximum (uniform per half).
__device__ __forceinline__ int argmax16(float v, unsigned shift) {
  float m = v;
#pragma unroll
  for (int s = 1; s < 16; s <<= 1)
    m = fmaxf(m, __shfl_xor(m, s, 32));           // masks < 16 stay in-half
  unsigned mask = (unsigned)(__ballot(v == m) >> shift) & 0xffffu;
  return __ffs(mask) - 1;                          // first max index
}

__global__ __launch_bounds__(THREADS) void byte_bitwise_ffn_kernel(
    const float* __restrict__ x, float* __restrict__ out, int nrows) {
  __shared__ float tile[ROWS_PER_BLOCK * D_FEAT];  // 8 KB tile

  const int t = threadIdx.x;
  const int baseRow = blockIdx.x * ROWS_PER_BLOCK;

  int validRows = nrows - baseRow;
  if (validRows <= 0) return;
  if (validRows > ROWS_PER_BLOCK) validRows = ROWS_PER_BLOCK;
  const uint32_t tileBytes = (uint32_t)validRows * D_FEAT * 4u;

  // ---- Stage 16 rows (8 KB) into LDS via CDNA5 async copies ----
  // 512 chunks of 16B; 256 threads issue 2 async b128 loads each.
  {
    const float* gbase = x + (size_t)baseRow * D_FEAT;
    const uint32_t ldsBase = (uint32_t)(uintptr_t)(&tile[0]);
    const uint32_t g0 = (uint32_t)t * 16u;
    const uint32_t g1 = g0 + 256u * 16u;
    const uint32_t l0 = ldsBase + g0;
    const uint32_t l1 = ldsBase + g1;
    if (g0 < tileBytes) {
      asm volatile("global_load_async_to_lds_b128 %0, %1, %2"
                   :: "v"(l0), "v"(g0), "s"(gbase) : "memory");
    }
    if (g1 < tileBytes) {
      asm volatile("global_load_async_to_lds_b128 %0, %1, %2"
                   :: "v"(l1), "v"(g1), "s"(gbase) : "memory");
    }
    asm volatile("s_wait_asynccnt 0" ::: "memory");
  }
  __syncthreads();

  // ---- Each wave32 handles 2 rows: lanes 0-15 -> row0, 16-31 -> row1 ----
  const int wave = t >> 5;
  const int lane = t & 31;
  const unsigned shift = (lane & 16);          // 0 or 16: half's ballot bits
  const int i = lane & 15;                     // element index in each field
  const int rowL = wave * 2 + (lane >> 4);     // 0..15 within tile
  const float* __restrict__ rowp = &tile[rowL * D_FEAT];

  // Field layout: ALU_LO=4, ALU_HI=20, AX_CARRY_LO=36, AX_CARRY_HI=52
  const float v0 = rowp[4 + i];
  const float v1 = rowp[20 + i];
  const float v2 = rowp[36 + i];
  const float v3 = rowp[52 + i];

  const int n0 = argmax16(v0, shift);
  const int n1 = argmax16(v1, shift);
  const int n2 = argmax16(v2, shift);
  const int n3 = argmax16(v3, shift);

  const int a = n0 | (n1 << 4);
  const int b = n2 | (n3 << 4);

  // Flags: one b128 LDS broadcast load (MARK_AX, OP_AND, OP_OR, OP_XOR)
  const float4 fl = *(const float4*)(rowp);
  const bool is_and = fl.y > 0.5f;
  const bool is_or  = fl.z > 0.5f;
  const bool is_xor = fl.w > 0.5f;

  // Priority AND > OR > XOR; tables are exactly bitwise ops on the bytes.
  const int res = is_and ? (a & b) : (is_or ? (a | b) : (a ^ b));
  const bool active = (fl.x >= 0.5f) && (is_and || is_or || is_xor);
  const float w = active ? 2.0f : 0.0f;

  // Target float4 chunks in the row: OUTPUT_LO=68 -> chunks 17..20,
  // OUTPUT_HI=84 -> chunks 21..24 (both in the lane's second chunk set).
  const int cl = 17 + ((res >> 2) & 3);   // chunk holding 68 + (res & 15)
  const int ch = 21 + ((res >> 6) & 3);   // chunk holding 84 + (res >> 4)

  if (baseRow + rowL < nrows) {
    // Copy-through: lane i handles chunks i and i+16 of its row.
    float4 c0v = *(const float4*)(rowp + i * 4);
    float4 c1v = *(const float4*)(rowp + (i + 16) * 4);

    const int chunk = i + 16;
    float* p = (float*)&c1v;
    if (chunk == cl) p[res & 3] += w;          // (68+lo) % 4 == lo & 3
    if (chunk == ch) p[(res >> 4) & 3] += w;   // (84+hi) % 4 == hi & 3

    float4* __restrict__ orow =
        (float4*)(out + (size_t)(baseRow + rowL) * D_FEAT);
    orow[i] = c0v;
    orow[i + 16] = c1v;
  }
}

extern "C" void kernel_launch(void* const* d_in, const int* in_sizes, int n_in,
                              void* d_out, int out_size, void* d_ws, size_t ws_size,
                              hipStream_t stream) {
  (void)n_in; (void)d_ws; (void)ws_size; (void)out_size;
  const float* x = (const float*)d_in[0];   // [B,S,128] fp32
  // d_in[1..3] (and/or/xor tables) unused: table[a][b] == float(a op b).
  float* out = (float*)d_out;

  const int nrows = in_sizes[0] / D_FEAT;   // B*S = 131072
  const int blocks = (nrows + ROWS_PER_BLOCK - 1) / ROWS_PER_BLOCK;
  hipLaunchKernelGGL(byte_bitwise_ffn_kernel, dim3(blocks), dim3(THREADS), 0,
                     stream, x, out, nrows);
}